// ProtienGAT_82042465288650
// MI455X (gfx1250) — compile-verified
//
#include <hip/hip_runtime.h>
#include <cmath>

typedef __attribute__((ext_vector_type(2))) float v2f;
typedef __attribute__((ext_vector_type(8))) float v8f;

#define NN 4096
#define KK 32
#define CC 128
#define VV 21

// workspace layout (float offsets)
#define WS_U    0
#define WS_ADDV 128
#define WS_OVEC 256
#define WS_G    384            // 21*128
#define WS_KEY  3072           // 4096
#define WS_BF   7168           // 8192 float2 = 16384 floats (B fragments)

// ---------------- precompute: u, addv, ovec, G ----------------
__global__ void kpre_vec(const float* __restrict__ bl1, const float* __restrict__ bo1,
                         const float* __restrict__ Wl2, const float* __restrict__ bl2,
                         const float* __restrict__ Wr2, const float* __restrict__ br2,
                         const float* __restrict__ be2, const float* __restrict__ bo2,
                         const float* __restrict__ Ws,  float* __restrict__ ws)
{
    int c = threadIdx.x;  // 0..127
    float u = 0.f, xr = 0.f;
    for (int i = 0; i < CC; ++i) {
        float c1 = bl1[i] + bo1[i];
        u  += c1 * Wl2[i*CC + c];
        xr += c1 * Wr2[i*CC + c];
    }
    ws[WS_U    + c] = u;
    ws[WS_ADDV + c] = xr + br2[c] + be2[c] + bl2[c];
    ws[WS_OVEC + c] = bl2[c] + bo2[c];
    for (int v = 0; v < VV; ++v) {
        float g = 0.f;
        for (int h = 0; h < CC; ++h) g += Ws[v*CC + h] * Wl2[(CC + h)*CC + c];
        ws[WS_G + v*CC + c] = g;
    }
}

// ---------------- precompute: decoding-order keys ----------------
__global__ void kpre_key(const float* __restrict__ mask, const float* __restrict__ chM,
                         const float* __restrict__ z, float* __restrict__ ws)
{
    int n = blockIdx.x * blockDim.x + threadIdx.x;
    if (n < NN) ws[WS_KEY + n] = (chM[n]*mask[n] + 1e-4f) * fabsf(z[n]);
}

// ---------------- precompute: We2 as WMMA B fragments ----------------
// Bf[((ct*32+kk)*32+lane)] = { We2[4kk+(lane<16?0:2)][16ct+(lane&15)],
//                              We2[4kk+(lane<16?1:3)][16ct+(lane&15)] }
__global__ void kpre_bf(const float* __restrict__ We2, float* __restrict__ ws)
{
    int t = blockIdx.x * blockDim.x + threadIdx.x;  // 0..8191
    if (t >= 8192) return;
    int lane = t & 31;
    int r   = 4*((t>>5)&31) + ((lane>>4)<<1);
    int col = 16*(t>>10) + (lane & 15);
    ws[WS_BF + 2*t + 0] = We2[(r    )*CC + col];
    ws[WS_BF + 2*t + 1] = We2[(r + 1)*CC + col];
}

// ---------------- fused main kernel ----------------
// block: 256 threads = 8 waves, 4 nodes (128 edges, 8 tiles of 16 edges)
__global__ __launch_bounds__(256) void kmain(
    const float* __restrict__ E, const int* __restrict__ Eidx, const int* __restrict__ S,
    const float* __restrict__ mask, const float* __restrict__ a2,
    const float* __restrict__ Wout, const float* __restrict__ bout,
    const float* __restrict__ ws, float* __restrict__ out)
{
    const float* u    = ws + WS_U;
    const float* addv = ws + WS_ADDV;
    const float* ovec = ws + WS_OVEC;
    const float* G    = ws + WS_G;
    const float* key  = ws + WS_KEY;
    const float* Bf   = ws + WS_BF;

    __shared__ float BfL[16384];    // 64KB: B fragments, LDS-resident
    __shared__ float sLDS[128];     // scores, then alpha*bw
    __shared__ float bwLDS[128];
    __shared__ int   sjLDS[128];
    __shared__ float outLDS[4][128];

    const int lane = threadIdx.x & 31;
    const int w    = threadIdx.x >> 5;
    const int nb   = blockIdx.x;
    const int e0   = nb*128 + w*16;          // first edge of this wave's tile

    // per-edge meta (lane L and L+16 both compute edge e0+(L&15))
    int   e_m = e0 + (lane & 15);
    int   n_m = e_m >> 5;
    int   j_m = Eidx[e_m];
    float kn  = key[n_m], kj = key[j_m];
    float mn  = mask[n_m];
    float bw_l = ((kn > kj) || (kn == kj && n_m > j_m)) ? mn : 0.f;
    int   sj_l = S[j_m];
    if (lane < 16) { bwLDS[w*16 + lane] = bw_l; sjLDS[w*16 + lane] = sj_l; }

    // hoist per-row scalars for the epilogue (row r -> edge e0+r / e0+8+r)
    float bw_r[8], mk_r[8]; int sj_r[8];
    const int sbase = (lane < 16) ? 0 : 8;
    #pragma unroll
    for (int r = 0; r < 8; ++r) {
        bw_r[r] = __shfl(bw_l, sbase + r, 32);
        mk_r[r] = __shfl(mn,   sbase + r, 32);
        sj_r[r] = __shfl(sj_l, sbase + r, 32);
    }

    // A fragments: 16x4 f32 layout, K chained over 32 steps (issue early)
    const float* Ab = E + (size_t)(e0 + (lane & 15))*CC + ((lane >> 4) << 1);
    v2f af[32];
    #pragma unroll
    for (int kk = 0; kk < 32; ++kk) af[kk] = *(const v2f*)(Ab + 4*kk);

    // cooperative stage of the 64KB B-fragment buffer into LDS (coalesced b128)
    {
        const float4* src = (const float4*)Bf;
        float4*       dst = (float4*)BfL;
        int t = threadIdx.x;
        #pragma unroll
        for (int q = 0; q < 16; ++q) dst[t + 256*q] = src[t + 256*q];
    }
    __syncthreads();

    v8f spart = {};
    #pragma unroll 1
    for (int ct = 0; ct < 8; ++ct) {
        const float* Bp = BfL + ct*2048 + lane*2;

        // Preload ALL 32 B fragments for this column tile, then fence the
        // scheduler so the ds_loads stay hoisted above the wmma chain:
        // waits become a pipelined dscnt countdown instead of 0-stalls.
        v2f bfr[32];
        #pragma unroll
        for (int kk = 0; kk < 32; ++kk) bfr[kk] = *(const v2f*)(Bp + kk*64);
        __builtin_amdgcn_sched_barrier(0);

        v8f acc = {};
        #pragma unroll
        for (int kk = 0; kk < 32; ++kk) {
            acc = __builtin_amdgcn_wmma_f32_16x16x4_f32(
                      false, af[kk], false, bfr[kk], (short)0, acc, false, false);
        }
        __builtin_amdgcn_sched_barrier(0);

        // fused epilogue: arg = ee + mask*u + bw*G[S_j] + (xr2+be2+bl2); s += lrelu(arg)*a2
        int c = ct*16 + (lane & 15);
        float uc = u[c], ad = addv[c], a2c = a2[c];
        #pragma unroll
        for (int r = 0; r < 8; ++r) {
            float g   = G[sj_r[r]*CC + c];
            float arg = acc[r] + mk_r[r]*uc + bw_r[r]*g + ad;
            float m   = (arg >= 0.f) ? arg : 0.2f*arg;
            spart[r] += m * a2c;
        }
    }

    // reduce partial dots over the 16 lanes of each half
    #pragma unroll
    for (int r = 0; r < 8; ++r) {
        float v = spart[r];
        v += __shfl_xor(v, 1, 32);
        v += __shfl_xor(v, 2, 32);
        v += __shfl_xor(v, 4, 32);
        v += __shfl_xor(v, 8, 32);
        if (lane == 0)  sLDS[w*16 + r]     = v;
        if (lane == 16) sLDS[w*16 + 8 + r] = v;
    }
    __syncthreads();

    // softmax + output + logits + log_softmax: wave w < 4 handles node nb*4+w
    if (w < 4) {
        int n = nb*4 + w;
        float s = sLDS[w*32 + lane];
        float mx = s;
        #pragma unroll
        for (int off = 16; off >= 1; off >>= 1) mx = fmaxf(mx, __shfl_xor(mx, off, 32));
        float ex = expf(s - mx);
        float sm = ex;
        #pragma unroll
        for (int off = 16; off >= 1; off >>= 1) sm += __shfl_xor(sm, off, 32);
        float abw = (ex / sm) * bwLDS[w*32 + lane];
        sLDS[w*32 + lane] = abw;

        float mnn = mask[n];
        int c4 = lane*4;
        float o0 = mnn*u[c4+0] + ovec[c4+0];
        float o1 = mnn*u[c4+1] + ovec[c4+1];
        float o2 = mnn*u[c4+2] + ovec[c4+2];
        float o3 = mnn*u[c4+3] + ovec[c4+3];
        for (int k = 0; k < 32; ++k) {
            float ab = sLDS[w*32 + k];
            const float* Gr = G + sjLDS[w*32 + k]*CC + c4;
            o0 += ab*Gr[0]; o1 += ab*Gr[1]; o2 += ab*Gr[2]; o3 += ab*Gr[3];
        }
        outLDS[w][c4+0] = o0; outLDS[w][c4+1] = o1;
        outLDS[w][c4+2] = o2; outLDS[w][c4+3] = o3;

        // logits: lane v (<21) computes out . W_out[:,v]
        int v = (lane < VV) ? lane : 0;
        float lg = bout[v];
        for (int c = 0; c < CC; ++c) lg += outLDS[w][c] * Wout[c*VV + v];
        float gx = (lane < VV) ? lg : -__builtin_inff();
        #pragma unroll
        for (int off = 16; off >= 1; off >>= 1) gx = fmaxf(gx, __shfl_xor(gx, off, 32));
        float eg = (lane < VV) ? expf(lg - gx) : 0.f;
        float zs = eg;
        #pragma unroll
        for (int off = 16; off >= 1; off >>= 1) zs += __shfl_xor(zs, off, 32);
        if (lane < VV) out[(size_t)n*VV + lane] = lg - gx - logf(zs);
    }
}

extern "C" void kernel_launch(void* const* d_in, const int* in_sizes, int n_in,
                              void* d_out, int out_size, void* d_ws, size_t ws_size,
                              hipStream_t stream)
{
    const float* E    = (const float*)d_in[0];
    const int*   Eidx = (const int*)  d_in[1];
    const int*   S    = (const int*)  d_in[2];
    const float* mask = (const float*)d_in[3];
    const float* chM  = (const float*)d_in[4];
    const float* z    = (const float*)d_in[5];
    const float* bl1  = (const float*)d_in[7];
    const float* bo1  = (const float*)d_in[13];
    const float* Wl2  = (const float*)d_in[14];
    const float* bl2  = (const float*)d_in[15];
    const float* Wr2  = (const float*)d_in[16];
    const float* br2  = (const float*)d_in[17];
    const float* We2  = (const float*)d_in[18];
    const float* be2  = (const float*)d_in[19];
    const float* a2   = (const float*)d_in[20];
    const float* bo2  = (const float*)d_in[21];
    const float* Ws_  = (const float*)d_in[22];
    const float* Wout = (const float*)d_in[23];
    const float* bout = (const float*)d_in[24];
    float* ws  = (float*)d_ws;
    float* out = (float*)d_out;

    hipLaunchKernelGGL(kpre_vec, dim3(1),  dim3(128), 0, stream,
                       bl1, bo1, Wl2, bl2, Wr2, br2, be2, bo2, Ws_, ws);
    hipLaunchKernelGGL(kpre_key, dim3(16), dim3(256), 0, stream, mask, chM, z, ws);
    hipLaunchKernelGGL(kpre_bf,  dim3(32), dim3(256), 0, stream, We2, ws);
    hipLaunchKernelGGL(kmain,    dim3(1024), dim3(256), 0, stream,
                       E, Eidx, S, mask, a2, Wout, bout, ws, out);
}